// GAT_74045236183294
// MI455X (gfx1250) — compile-verified
//
#include <hip/hip_runtime.h>
#include <hip/hip_bf16.h>

#define HEADS 4
#define OUT_CH 16
#define GN 64          // HEADS*OUT_CH
#define IN_CH 128
#define NEG_SLOPE 0.2f

typedef __attribute__((ext_vector_type(2))) float v2f;
typedef __attribute__((ext_vector_type(8))) float v8f;

// ---------------------------------------------------------------------------
// h = x @ weight   (N x 128) @ (128 x 64)  via V_WMMA_F32_16X16X4_F32.
// One wave per 16x16 output tile; K-loop of 32 WMMA steps (K=4 each).
// A 16x4 f32 layout: lanes 0-15 -> M=lane, VGPR0=K(2*half), VGPR1=K(2*half+1).
// B 4x16 f32 layout: lanes hold N=lane%16, VGPR0=K(2*half), VGPR1=K(2*half+1).
// C/D 16x16 f32: VGPR r -> M = r + 8*half, N = lane%16.
// ---------------------------------------------------------------------------
__global__ __launch_bounds__(256) void gat_gemm_wmma(
    const float* __restrict__ x, const float* __restrict__ w,
    float* __restrict__ h, int mtiles) {
  const int wave = (int)((blockIdx.x * blockDim.x + threadIdx.x) >> 5);
  const int lane = threadIdx.x & 31;
  const int ntiles = GN / 16;  // 4
  if (wave >= mtiles * ntiles) return;  // uniform per wave: EXEC stays all-1s
  const int tm = wave / ntiles;
  const int tn = wave % ntiles;
  const int m0 = tm * 16;
  const int n0 = tn * 16;
  const int half = lane >> 4;    // 0 or 1
  const int lrow = lane & 15;

  const float* arow = x + (size_t)(m0 + lrow) * IN_CH;
  v8f c = {};
#pragma unroll
  for (int k0 = 0; k0 < IN_CH; k0 += 4) {
    v2f a = *(const v2f*)(arow + k0 + 2 * half);        // contiguous pair in row
    v2f b;
    b.x = w[(size_t)(k0 + 2 * half) * GN + n0 + lrow];
    b.y = w[(size_t)(k0 + 2 * half + 1) * GN + n0 + lrow];
    // 8 args: (neg_a, A, neg_b, B, c_mod, C, reuse_a, reuse_b)
    c = __builtin_amdgcn_wmma_f32_16x16x4_f32(false, a, false, b,
                                              (short)0, c, false, false);
  }
#pragma unroll
  for (int r = 0; r < 8; ++r) {
    h[(size_t)(m0 + r + 8 * half) * GN + n0 + lrow] = c[r];
  }
}

// Scalar tail for rows not covered by 16-row tiles (unused when N%16==0).
__global__ void gat_gemm_tail(const float* __restrict__ x,
                              const float* __restrict__ w,
                              float* __restrict__ h, int mstart, int N) {
  int i = (int)(blockIdx.x * blockDim.x + threadIdx.x);
  int row = mstart + i / GN;
  int cc = i % GN;
  if (row >= N) return;
  float acc = 0.f;
  for (int k = 0; k < IN_CH; ++k)
    acc += x[(size_t)row * IN_CH + k] * w[(size_t)k * GN + cc];
  h[(size_t)row * GN + cc] = acc;
}

// out[n, c] = bias[c]  (scatter adds land on top)
__global__ void gat_init_out(float* __restrict__ out,
                             const float* __restrict__ bias, int total) {
  int i = (int)(blockIdx.x * blockDim.x + threadIdx.x);
  if (i < total) out[i] = bias[i & (GN - 1)];
}

// ---------------------------------------------------------------------------
// Edge phase: one thread per (edge, head). 4 contiguous lanes form one edge;
// softmax over the 4 heads via __shfl_xor within the aligned 4-lane group.
// Gathers are b128 (float4) from L2-resident h; scatter is atomic f32 adds
// into L2-resident out (collisions required by semantics).
// ---------------------------------------------------------------------------
__global__ __launch_bounds__(256) void gat_edge(
    const int* __restrict__ ei, const float* __restrict__ h,
    const float* __restrict__ att, float* __restrict__ out, int N, int E) {
  int gid = (int)(blockIdx.x * blockDim.x + threadIdx.x);
  int total = (E + N) * HEADS;
  if (gid >= total) return;
  int e = gid >> 2;          // edge id (incl. self loops)
  int hd = gid & 3;          // head id
  int row, col;
  if (e < E) {
    row = ei[e];
    col = ei[E + e];
  } else {                   // self loop: row = col = node id
    row = e - E;
    col = row;
  }

  const float4* xi = (const float4*)(h + (size_t)row * GN + hd * OUT_CH);
  const float4* xj = (const float4*)(h + (size_t)col * GN + hd * OUT_CH);
  const float4* ai = (const float4*)(att + hd * 2 * OUT_CH);            // att_i
  const float4* aj = (const float4*)(att + hd * 2 * OUT_CH + OUT_CH);   // att_j

  float4 xiv[4];
  float alpha = 0.f;
#pragma unroll
  for (int q = 0; q < 4; ++q) {
    xiv[q] = xi[q];
    float4 xjv = xj[q];
    float4 a = ai[q];
    float4 b = aj[q];
    alpha += xiv[q].x * a.x + xiv[q].y * a.y + xiv[q].z * a.z + xiv[q].w * a.w;
    alpha += xjv.x * b.x + xjv.y * b.y + xjv.z * b.z + xjv.w * b.w;
  }
  alpha = (alpha >= 0.f) ? alpha : NEG_SLOPE * alpha;   // leaky_relu

  // softmax over the 4 heads of this edge (lanes 4k..4k+3 of a wave32)
  float m = alpha;
  m = fmaxf(m, __shfl_xor(m, 1, 32));
  m = fmaxf(m, __shfl_xor(m, 2, 32));
  float ex = __expf(alpha - m);
  float s = ex + __shfl_xor(ex, 1, 32);
  s = s + __shfl_xor(s, 2, 32);
  float wgt = ex / s;

  float* op = out + (size_t)col * GN + hd * OUT_CH;
#pragma unroll
  for (int q = 0; q < 4; ++q) {
    atomicAdd(op + 4 * q + 0, xiv[q].x * wgt);
    atomicAdd(op + 4 * q + 1, xiv[q].y * wgt);
    atomicAdd(op + 4 * q + 2, xiv[q].z * wgt);
    atomicAdd(op + 4 * q + 3, xiv[q].w * wgt);
  }
}

extern "C" void kernel_launch(void* const* d_in, const int* in_sizes, int n_in,
                              void* d_out, int out_size, void* d_ws, size_t ws_size,
                              hipStream_t stream) {
  const float* x    = (const float*)d_in[0];   // (N, 128)
  const int*   ei   = (const int*)d_in[1];     // (2, E)
  const float* w    = (const float*)d_in[2];   // (128, 64)
  const float* att  = (const float*)d_in[3];   // (1, 4, 32)
  const float* bias = (const float*)d_in[4];   // (64,)
  float* out = (float*)d_out;                  // (N, 64)
  float* h = (float*)d_ws;                     // scratch: N*64 floats

  const int N = in_sizes[0] / IN_CH;           // 50000
  const int E = in_sizes[1] / 2;               // 800000

  // 1) GEMM: h = x @ w  (WMMA f32 16x16x4)
  const int mtiles = N / 16;
  const int waves = mtiles * (GN / 16);
  const int gthreads = waves * 32;
  if (gthreads > 0)
    gat_gemm_wmma<<<(gthreads + 255) / 256, 256, 0, stream>>>(x, w, h, mtiles);
  const int mrem = N - mtiles * 16;
  if (mrem > 0) {
    int t = mrem * GN;
    gat_gemm_tail<<<(t + 255) / 256, 256, 0, stream>>>(x, w, h, mtiles * 16, N);
  }

  // 2) out = bias (broadcast)
  const int ototal = N * GN;
  gat_init_out<<<(ototal + 255) / 256, 256, 0, stream>>>(out, bias, ototal);

  // 3) edge gather + per-edge head-softmax + atomic scatter-add
  const int etotal = (E + N) * HEADS;
  gat_edge<<<(etotal + 255) / 256, 256, 0, stream>>>(ei, h, att, out, N, E);
}